// RGCF_31628139168298
// MI455X (gfx1250) — compile-verified
//
#include <hip/hip_runtime.h>

#define N_USERS 40000
#define N_ITEMS 20000
#define NTOT    60000          // N_USERS + N_ITEMS
#define NNZ     1000000
#define DDIM    64
#define EMBN    (NTOT * DDIM)  // 3,840,000 floats

typedef __attribute__((ext_vector_type(2))) float v2f;
typedef __attribute__((ext_vector_type(8))) float v8f;

// ---------------------------------------------------------------------------
// K1: emb0[u]      += user_linear[i]   (user part, rows 0..N_USERS-1)
//     emb0[NU + i] += item_linear[u]   (item part, rows N_USERS..NTOT-1)
// one thread per (edge, 4-float chunk): 16M threads
// ---------------------------------------------------------------------------
__global__ void rgcf_scatter_emb(const float* __restrict__ user_linear,
                                 const float* __restrict__ item_linear,
                                 const int* __restrict__ u_idx,
                                 const int* __restrict__ i_idx,
                                 float* __restrict__ emb) {
  int t = blockIdx.x * blockDim.x + threadIdx.x;       // < NNZ*16 exactly
  int e = t >> 4;
  int q = (t & 15) * 4;
  int u  = u_idx[e];
  int it = i_idx[e];
  float4 a = *(const float4*)(user_linear + (size_t)it * DDIM + q);
  float* du = emb + (size_t)u * DDIM + q;
  atomicAdd(du + 0, a.x); atomicAdd(du + 1, a.y);
  atomicAdd(du + 2, a.z); atomicAdd(du + 3, a.w);
  float4 b = *(const float4*)(item_linear + (size_t)u * DDIM + q);
  float* di = emb + (size_t)(N_USERS + it) * DDIM + q;
  atomicAdd(di + 0, b.x); atomicAdd(di + 1, b.y);
  atomicAdd(di + 2, b.z); atomicAdd(di + 3, b.w);
}

// ---------------------------------------------------------------------------
// K2: rn[r] = 1 / max(||emb[r]||, 1e-8)
// ---------------------------------------------------------------------------
__global__ void rgcf_rownorm(const float* __restrict__ emb, float* __restrict__ rn) {
  int r = blockIdx.x * blockDim.x + threadIdx.x;
  if (r >= NTOT) return;
  const float* row = emb + (size_t)r * DDIM;
  float s = 0.0f;
#pragma unroll
  for (int k = 0; k < DDIM; k += 4) {
    float4 v = *(const float4*)(row + k);
    s += v.x * v.x + v.y * v.y + v.z * v.z + v.w * v.w;
  }
  rn[r] = 1.0f / fmaxf(sqrtf(s), 1e-8f);
}

// ---------------------------------------------------------------------------
// K3: per-edge cosine similarity via V_WMMA_F32_16X16X4_F32.
// One wave32 handles 16 edges. A = 16 user rows, B = 16 item rows^T,
// chained over K=64 in 16 steps of K=4; C diagonal = the 16 dot products.
//   32-bit A 16x4 layout : lane L holds A[M=L&15][K = 4kc + 2*(L>>4) + {0,1}]
//   32-bit B 4x16 layout : lane L holds B[K = 4kc + 2*(L>>4) + {0,1}][N=L&15]
//   C 16x16 layout       : diag m<8 -> lane m, c[m]; diag m>=8 -> lane m+16, c[m-8]
// sim = dot_raw * rn[u] * rn[NU+i];  sv = (sim+1)/2;  diag[u]+=sv; diag[NU+i]+=sv
// ---------------------------------------------------------------------------
__global__ void rgcf_sim_wmma(const int* __restrict__ u_idx,
                              const int* __restrict__ i_idx,
                              const float* __restrict__ emb,
                              const float* __restrict__ rn,
                              float* __restrict__ sv,
                              float* __restrict__ diag) {
  const int lane = threadIdx.x & 31;
  const int wave = (blockIdx.x * (blockDim.x >> 5)) + (threadIdx.x >> 5);
  const int e0   = wave * 16;                 // grid sized so e0 < NNZ always

  const int m  = lane & 15;
  const int hi = lane >> 4;
  const int e  = e0 + m;
  const int u  = u_idx[e];
  const int it = i_idx[e];
  const float* urow = emb + (size_t)u * DDIM;
  const float* irow = emb + (size_t)(N_USERS + it) * DDIM;

  v8f c = {};
#pragma unroll
  for (int kc = 0; kc < 16; ++kc) {
    const int off = kc * 4 + hi * 2;
    v2f a; a[0] = urow[off]; a[1] = urow[off + 1];
    v2f b; b[0] = irow[off]; b[1] = irow[off + 1];
    c = __builtin_amdgcn_wmma_f32_16x16x4_f32(
        /*neg_a=*/false, a, /*neg_b=*/false, b,
        /*c_mod=*/(short)0, c, /*reuse_a=*/false, /*reuse_b=*/false);
  }

  // extract diagonal: lanes 0..7 own edges 0..7 (c[lane]); lanes 24..31 own
  // edges 8..15 (c[lane-24]). In both cases index = lane & 7, edge = lane & 15.
  const int cidx = lane & 7;
  float d = c[0];
#pragma unroll
  for (int k = 1; k < 8; ++k) d = (cidx == k) ? c[k] : d;

  if ((lane < 8) || (lane >= 24)) {
    float sim = d * rn[u] * rn[N_USERS + it];
    float s = (sim + 1.0f) * 0.5f;
    sv[e] = s;
    atomicAdd(&diag[u], s);
    atomicAdd(&diag[N_USERS + it], s);
  }
}

// ---------------------------------------------------------------------------
// K4: per-edge propagation weights (fixed across layers)
//   w1[e] = sv[e] / (diag[u]+1e-7)      (user <- item direction)
//   w2[e] = sv[e] / (diag[NU+i]+1e-7)   (item <- user direction)
// ---------------------------------------------------------------------------
__global__ void rgcf_weights(const int* __restrict__ u_idx,
                             const int* __restrict__ i_idx,
                             const float* __restrict__ sv,
                             const float* __restrict__ diag,
                             float* __restrict__ w1,
                             float* __restrict__ w2) {
  int e = blockIdx.x * blockDim.x + threadIdx.x;
  if (e >= NNZ) return;
  float s = sv[e];
  w1[e] = s / (diag[u_idx[e]] + 1e-7f);
  w2[e] = s / (diag[N_USERS + i_idx[e]] + 1e-7f);
}

// ---------------------------------------------------------------------------
// K5: one propagation layer: nxt[u] += w1*cur[NU+i]; nxt[NU+i] += w2*cur[u]
// one thread per (edge, 4-float chunk)
// ---------------------------------------------------------------------------
__global__ void rgcf_propagate(const int* __restrict__ u_idx,
                               const int* __restrict__ i_idx,
                               const float* __restrict__ w1,
                               const float* __restrict__ w2,
                               const float* __restrict__ cur,
                               float* __restrict__ nxt) {
  int t = blockIdx.x * blockDim.x + threadIdx.x;       // < NNZ*16 exactly
  int e = t >> 4;
  int q = (t & 15) * 4;
  int u  = u_idx[e];
  int it = i_idx[e];
  float a = w1[e];
  float b = w2[e];
  float4 si = *(const float4*)(cur + (size_t)(N_USERS + it) * DDIM + q);
  float* du = nxt + (size_t)u * DDIM + q;
  atomicAdd(du + 0, a * si.x); atomicAdd(du + 1, a * si.y);
  atomicAdd(du + 2, a * si.z); atomicAdd(du + 3, a * si.w);
  float4 su = *(const float4*)(cur + (size_t)u * DDIM + q);
  float* di = nxt + (size_t)(N_USERS + it) * DDIM + q;
  atomicAdd(di + 0, b * su.x); atomicAdd(di + 1, b * su.y);
  atomicAdd(di + 2, b * su.z); atomicAdd(di + 3, b * su.w);
}

// ---------------------------------------------------------------------------
// accumulation helpers (float4, EMBN divisible by 4)
// ---------------------------------------------------------------------------
__global__ void rgcf_assign4(float* __restrict__ out, const float* __restrict__ a) {
  int t = blockIdx.x * blockDim.x + threadIdx.x;
  ((float4*)out)[t] = ((const float4*)a)[t];
}
__global__ void rgcf_add4(float* __restrict__ out, const float* __restrict__ a) {
  int t = blockIdx.x * blockDim.x + threadIdx.x;
  float4 o = ((float4*)out)[t];
  float4 v = ((const float4*)a)[t];
  o.x += v.x; o.y += v.y; o.z += v.z; o.w += v.w;
  ((float4*)out)[t] = o;
}
__global__ void rgcf_add_scale4(float* __restrict__ out, const float* __restrict__ a,
                                float sc) {
  int t = blockIdx.x * blockDim.x + threadIdx.x;
  float4 o = ((float4*)out)[t];
  float4 v = ((const float4*)a)[t];
  o.x = (o.x + v.x) * sc; o.y = (o.y + v.y) * sc;
  o.z = (o.z + v.z) * sc; o.w = (o.w + v.w) * sc;
  ((float4*)out)[t] = o;
}

// ---------------------------------------------------------------------------
extern "C" void kernel_launch(void* const* d_in, const int* in_sizes, int n_in,
                              void* d_out, int out_size, void* d_ws, size_t ws_size,
                              hipStream_t stream) {
  const float* user_linear = (const float*)d_in[0];   // [N_ITEMS, 64]
  const float* item_linear = (const float*)d_in[1];   // [N_USERS, 64]
  const int*   u_idx       = (const int*)d_in[2];     // [NNZ]
  const int*   i_idx       = (const int*)d_in[3];     // [NNZ]
  float*       out         = (float*)d_out;           // [NTOT, 64]

  float* ws   = (float*)d_ws;
  float* emb0 = ws;                 // EMBN
  float* bufA = emb0 + EMBN;        // EMBN
  float* bufB = bufA + EMBN;        // EMBN
  float* sv   = bufB + EMBN;        // NNZ
  float* w1   = sv   + NNZ;         // NNZ
  float* w2   = w1   + NNZ;         // NNZ
  float* diag = w2   + NNZ;         // NTOT
  float* rn   = diag + NTOT;        // NTOT

  const int edgechunk_blocks = (NNZ * 16) / 256;      // 62500, exact
  const int n4_blocks        = (EMBN / 4) / 256;      // 3750, exact
  const int sim_blocks       = (NNZ / 16) / 4;        // 15625 blocks of 4 waves

  // ---- embeddings via scatter ----
  hipMemsetAsync(emb0, 0, (size_t)EMBN * 4, stream);
  hipMemsetAsync(diag, 0, (size_t)NTOT * 4, stream);
  rgcf_scatter_emb<<<edgechunk_blocks, 256, 0, stream>>>(user_linear, item_linear,
                                                         u_idx, i_idx, emb0);
  // ---- inverse row norms ----
  rgcf_rownorm<<<(NTOT + 255) / 256, 256, 0, stream>>>(emb0, rn);
  // ---- WMMA cosine sims + degree accumulation ----
  rgcf_sim_wmma<<<sim_blocks, 128, 0, stream>>>(u_idx, i_idx, emb0, rn, sv, diag);
  // ---- per-edge normalized weights ----
  rgcf_weights<<<(NNZ + 255) / 256, 256, 0, stream>>>(u_idx, i_idx, sv, diag, w1, w2);

  // ---- mean accumulator starts at layer-0 embedding ----
  rgcf_assign4<<<n4_blocks, 256, 0, stream>>>(out, emb0);

  // layer 1: emb0 -> bufA
  hipMemsetAsync(bufA, 0, (size_t)EMBN * 4, stream);
  rgcf_propagate<<<edgechunk_blocks, 256, 0, stream>>>(u_idx, i_idx, w1, w2, emb0, bufA);
  rgcf_add4<<<n4_blocks, 256, 0, stream>>>(out, bufA);

  // layer 2: bufA -> bufB
  hipMemsetAsync(bufB, 0, (size_t)EMBN * 4, stream);
  rgcf_propagate<<<edgechunk_blocks, 256, 0, stream>>>(u_idx, i_idx, w1, w2, bufA, bufB);
  rgcf_add4<<<n4_blocks, 256, 0, stream>>>(out, bufB);

  // layer 3: bufB -> bufA (reuse), then out = (out + bufA) / 4
  hipMemsetAsync(bufA, 0, (size_t)EMBN * 4, stream);
  rgcf_propagate<<<edgechunk_blocks, 256, 0, stream>>>(u_idx, i_idx, w1, w2, bufB, bufA);
  rgcf_add_scale4<<<n4_blocks, 256, 0, stream>>>(out, bufA, 0.25f);
}